// CGCL_60567628808330
// MI455X (gfx1250) — compile-verified
//
#include <hip/hip_runtime.h>
#include <hip/hip_bf16.h>
#include <math.h>

#define FIN  128
#define FHID 64

typedef float v2f __attribute__((ext_vector_type(2)));
typedef float v8f __attribute__((ext_vector_type(8)));

// ---------------------------------------------------------------------------
// Phase 1: degree of each destination node (integer atomics, L2-native).
// ---------------------------------------------------------------------------
__global__ void cgcl_deg_kernel(const int* __restrict__ dst,
                                unsigned* __restrict__ deg, int E) {
    int e = blockIdx.x * blockDim.x + threadIdx.x;
    if (e < E) atomicAdd(&deg[dst[e]], 1u);
}

// Phase 2: dinv[n] = rsqrt(deg[n] + 1)  (self-loop included)
__global__ void cgcl_dinv_kernel(const unsigned* __restrict__ deg,
                                 float* __restrict__ dinv, int N) {
    int n = blockIdx.x * blockDim.x + threadIdx.x;
    if (n < N) dinv[n] = rsqrtf((float)deg[n] + 1.0f);
}

// ---------------------------------------------------------------------------
// Phase 3: h = x @ W via V_WMMA_F32_16X16X4_F32.
// One wave = one 16x16 fp32 tile; 4 waves/block cover FHID=64 columns of a
// 16-row tile; 32 chained WMMAs walk K=128.
//
// A (16x4 f32) layout: lane (m = lane&15) holds row m; VGPR0/1 hold
//   K = koff / koff+1 with koff = (lane<16 ? 0 : 2).
// B (4x16 f32) layout: lane&15 = column; VGPR0/1 = K parity; lane-half picks
//   K base 0 or 2 (mirrors the A striping).
// C/D (16x16 f32): VGPR v -> row v + (lane<16?0:8), col = lane&15.
// ---------------------------------------------------------------------------
__global__ void __launch_bounds__(128)
cgcl_gemm_wmma(const float* __restrict__ x, const float* __restrict__ W,
               float* __restrict__ h, int N) {
    const int lane = threadIdx.x & 31;
    const int wave = threadIdx.x >> 5;        // column tile 0..3
    const int m    = lane & 15;
    const int hi   = lane >> 4;               // 0: K pair {0,1}, 1: {2,3}
    const int koff = hi * 2;

    int row = blockIdx.x * 16 + m;
    if (row >= N) row = N - 1;                // keep EXEC all-ones for WMMA
    const int col = wave * 16 + m;

    const float* __restrict__ xrow = x + (size_t)row * FIN;

    v8f c = {};
#pragma unroll 4
    for (int kk = 0; kk < FIN / 4; ++kk) {
        const int k = kk * 4 + koff;
        v2f a = *(const v2f*)(xrow + k);      // 8B-aligned (k even)
        v2f bfrag;
        bfrag.x = W[(k + 0) * FHID + col];
        bfrag.y = W[(k + 1) * FHID + col];
        c = __builtin_amdgcn_wmma_f32_16x16x4_f32(
                /*neg_a=*/false, a, /*neg_b=*/false, bfrag,
                /*c_mod=*/(short)0, c, /*reuse_a=*/false, /*reuse_b=*/false);
    }

    const int mbase = blockIdx.x * 16 + hi * 8;
    float* __restrict__ hout = h + (size_t)mbase * FHID + col;
    if (blockIdx.x * 16 + 16 <= N) {
        // Full tile (always true when N % 16 == 0): unguarded store clause,
        // no exec-mask churn.
#pragma unroll
        for (int v = 0; v < 8; ++v) {
            hout[(size_t)v * FHID] = c[v];
        }
    } else {
        // Generic tail tile (not hit at N=100000).
#pragma unroll
        for (int v = 0; v < 8; ++v) {
            if (mbase + v < N) hout[(size_t)v * FHID] = c[v];
        }
    }
}

// ---------------------------------------------------------------------------
// Phase 4: edge scatter  out[dst] += h[src] * dinv[src]*dinv[dst].
// 16 threads per edge, each handles 4 features (float4 gather + 4 native
// fp32 atomics). Dominant phase; h fits in the 192MB L2 so the degree-16
// reuse of h[src] stays on-chip.
// ---------------------------------------------------------------------------
__global__ void cgcl_scatter_kernel(const int* __restrict__ src,
                                    const int* __restrict__ dst,
                                    const float* __restrict__ dinv,
                                    const float* __restrict__ h,
                                    float* __restrict__ out, int E) {
    int gid = blockIdx.x * blockDim.x + threadIdx.x;
    int e = gid >> 4;
    if (e >= E) return;
    int q = gid & 15;
    int s = src[e];
    int d = dst[e];
    float cf = dinv[s] * dinv[d];
    const float4 hv = *(const float4*)(h + (size_t)s * FHID + q * 4);
    float* o = out + (size_t)d * FHID + q * 4;
    unsafeAtomicAdd(o + 0, hv.x * cf);   // global_atomic_add_f32
    unsafeAtomicAdd(o + 1, hv.y * cf);
    unsafeAtomicAdd(o + 2, hv.z * cf);
    unsafeAtomicAdd(o + 3, hv.w * cf);
}

// ---------------------------------------------------------------------------
// Phase 5: out = ELU(agg + h * dinv^2 + b)   (self-loop + bias + activation)
// ---------------------------------------------------------------------------
__global__ void cgcl_finalize_kernel(const float* __restrict__ h,
                                     const float* __restrict__ dinv,
                                     const float* __restrict__ bias,
                                     float* __restrict__ out, int total) {
    int gid = blockIdx.x * blockDim.x + threadIdx.x;
    if (gid >= total) return;
    int n = gid >> 6;          // / FHID
    int f = gid & (FHID - 1);
    float di = dinv[n];
    float v = out[gid] + h[gid] * di * di + bias[f];
    out[gid] = (v > 0.0f) ? v : expm1f(v);
}

extern "C" void kernel_launch(void* const* d_in, const int* in_sizes, int n_in,
                              void* d_out, int out_size, void* d_ws, size_t ws_size,
                              hipStream_t stream) {
    const float* x    = (const float*)d_in[0];   // [N, 128]
    const float* W    = (const float*)d_in[1];   // [128, 64]
    const float* bias = (const float*)d_in[2];   // [64]
    const int*   ei   = (const int*)d_in[3];     // [2, E] row-major

    const int N = in_sizes[0] / FIN;
    const int E = in_sizes[3] / 2;
    const int* src = ei;
    const int* dst = ei + E;

    // Workspace layout (4-byte units): deg[N] (u32) | dinv[N] (f32) | h[N*64]
    size_t NA = ((size_t)N + 255) & ~(size_t)255;
    unsigned* deg  = (unsigned*)d_ws;
    float*    dinv = (float*)d_ws + NA;
    float*    h    = (float*)d_ws + 2 * NA;
    float*    out  = (float*)d_out;

    // Zero accumulators each call (graph replays don't re-poison).
    hipMemsetAsync(deg, 0, (size_t)N * sizeof(unsigned), stream);
    hipMemsetAsync(out, 0, (size_t)out_size * sizeof(float), stream);

    cgcl_deg_kernel<<<(E + 255) / 256, 256, 0, stream>>>(dst, deg, E);
    cgcl_dinv_kernel<<<(N + 255) / 256, 256, 0, stream>>>(deg, dinv, N);

    const int row_tiles = (N + 15) / 16;
    cgcl_gemm_wmma<<<row_tiles, 128, 0, stream>>>(x, W, h, N);

    const long long sc_threads = (long long)E * 16;
    cgcl_scatter_kernel<<<(unsigned)((sc_threads + 255) / 256), 256, 0, stream>>>(
        src, dst, dinv, h, out, E);

    const int total = N * FHID;
    cgcl_finalize_kernel<<<(total + 255) / 256, 256, 0, stream>>>(
        h, dinv, bias, out, total);
}